// MultiTurnFeatureAwareCRF_8564164788627
// MI455X (gfx1250) — compile-verified
//
#include <hip/hip_runtime.h>
#include <stdint.h>

// Problem constants (match reference: B=4096, S=2048, T=2)
static constexpr int BB      = 4096;
static constexpr int SS      = 2048;
static constexpr int NCHUNK  = 32;
static constexpr int CHUNK   = SS / NCHUNK;   // 64 steps per chunk
static constexpr float NEGI  = -1.0e30f;      // finite "-inf" (keeps lse2 NaN-free)

// Clang ext-vector types (HIP's int4/float4 are structs; the nontemporal
// builtin requires true vector types).
typedef int   vi4 __attribute__((ext_vector_type(4)));
typedef float vf4 __attribute__((ext_vector_type(4)));

// log(exp(a)+exp(b)) using hardware v_exp_f32/v_log_f32 (short dependent chain).
// Exact when one arg is ~-1e30: exp(-huge)=0 -> log(1)=0.
__device__ __forceinline__ float lse2(float a, float b) {
    float m = fmaxf(a, b);
    float d = fabsf(a - b);
    return m + __logf(1.0f + __expf(-d));
}

// ---------------------------------------------------------------------------
// Kernel 1: one thread per (batch, chunk). Streams 64 steps (NT-hinted, b128
// loads), builds the chunk's 2x2 log-semiring transfer matrix (two row
// recursions) + partial gold score. Transition bank lives in LDS, populated
// via the CDNA5 async global->LDS path.
// ---------------------------------------------------------------------------
__global__ __launch_bounds__(256) void crf_chunk_kernel(
    const float* __restrict__ em,     // [B,S,2]
    const float* __restrict__ tp,     // [4,2,2]
    const int*   __restrict__ tags,   // [B,S]
    const int*   __restrict__ w2w,    // [B,S]
    const int*   __restrict__ itm,    // [B,S]
    const int*   __restrict__ dst,    // [B,S]
    float*       __restrict__ ws)     // [NCHUNK,5,B]
{
    __shared__ alignas(16) float bank[32];    // [5][4]: sel 0 = zeros, 1..4 = tp
    const int tid = threadIdx.x;

    if (tid < 4) bank[tid] = 0.0f;            // padding matrix (who2who == -1)
    // CDNA5 async global->LDS load of the 16 learned transition params.
    if (tid < 16) {
        unsigned lds_off = (unsigned)(uintptr_t)(&bank[4 + tid]);
        unsigned long long ga = (unsigned long long)(uintptr_t)(tp + tid);
        asm volatile("global_load_async_to_lds_b32 %0, %1, off"
                     :: "v"(lds_off), "v"(ga) : "memory");
    }
    asm volatile("s_wait_asynccnt 0" ::: "memory");
    __syncthreads();

    const int b  = blockIdx.x * blockDim.x + tid;   // batch index
    const int c  = blockIdx.y;                      // chunk index
    const int s0 = c * CHUNK;
    const long base = (long)b * SS;

    const vf4* em4 = (const vf4*)(em + (base + s0) * 2); // 2 steps / vf4
    const vi4* tg4 = (const vi4*)(tags + base + s0);     // 4 steps / vi4
    const vi4* ww4 = (const vi4*)(w2w  + base + s0);
    const vi4* it4 = (const vi4*)(itm  + base + s0);
    const vi4* dd4 = (const vi4*)(dst  + base + s0);

    // Chunk transfer matrix as two row recursions (log-semiring identity init)
    float a00 = 0.0f, a01 = NEGI, a10 = NEGI, a11 = 0.0f;
    float gold = 0.0f;
    int ptag = (s0 > 0) ? tags[base + s0 - 1] : 0;

    auto step = [&](float e0, float e1, int tg_, int ww_, int it_, int dd_, int sg) {
        int sel = (ww_ == -1) ? 0 : (ww_ == 1) ? 1 : (it_ == 0) ? 2 : (dd_ == 0) ? 3 : 4;
        vf4 tm = *(const vf4*)&bank[sel * 4];   // ds_load_b128, dynamic sel
        // row_i[j] <- em[j] + lse_k(row_i[k] + tr[k][j])
        float n00 = e0 + lse2(a00 + tm.x, a01 + tm.z);
        float n01 = e1 + lse2(a00 + tm.y, a01 + tm.w);
        float n10 = e0 + lse2(a10 + tm.x, a11 + tm.z);
        float n11 = e1 + lse2(a10 + tm.y, a11 + tm.w);
        a00 = n00; a01 = n01; a10 = n10; a11 = n11;
        // gold path: emission + (transition for global step > 0)
        gold += tg_ ? e1 : e0;
        float tv = ptag ? (tg_ ? tm.w : tm.z) : (tg_ ? tm.y : tm.x);
        gold += (sg == 0) ? 0.0f : tv;
        ptag = tg_;
    };

    constexpr int Q = CHUNK / 4;   // 16 quad-step iterations
    #pragma unroll 2
    for (int q = 0; q < Q; ++q) {
        if (((q & 7) == 0) && (q + 8 < Q)) {         // stream prefetch, 1 line ahead
            __builtin_prefetch(&em4[2 * q + 8], 0, 0);
            __builtin_prefetch(&tg4[q + 8], 0, 0);
            __builtin_prefetch(&ww4[q + 8], 0, 0);
            __builtin_prefetch(&it4[q + 8], 0, 0);
            __builtin_prefetch(&dd4[q + 8], 0, 0);
        }
        // Streamed exactly once across the whole kernel -> non-temporal hints
        // (keep the 192MB L2 clean; total stream is ~201MB with zero reuse).
        vi4 tg = __builtin_nontemporal_load(&tg4[q]);
        vi4 ww = __builtin_nontemporal_load(&ww4[q]);
        vi4 it = __builtin_nontemporal_load(&it4[q]);
        vi4 dd = __builtin_nontemporal_load(&dd4[q]);
        vf4 eA = __builtin_nontemporal_load(&em4[2 * q]);     // steps 4q, 4q+1
        vf4 eB = __builtin_nontemporal_load(&em4[2 * q + 1]); // steps 4q+2, 4q+3
        int sg = s0 + 4 * q;
        step(eA.x, eA.y, tg.x, ww.x, it.x, dd.x, sg + 0);
        step(eA.z, eA.w, tg.y, ww.y, it.y, dd.y, sg + 1);
        step(eB.x, eB.y, tg.z, ww.z, it.z, dd.z, sg + 2);
        step(eB.z, eB.w, tg.w, ww.w, it.w, dd.w, sg + 3);
    }

    // Workspace layout [c][field][b] -> coalesced stores now, coalesced loads later
    float* o = ws + (size_t)c * 5 * BB;
    o[0 * BB + b] = a00;
    o[1 * BB + b] = a01;
    o[2 * BB + b] = a10;
    o[3 * BB + b] = a11;
    o[4 * BB + b] = gold;
}

// ---------------------------------------------------------------------------
// Kernel 2: one thread per batch. Composes the NCHUNK chunk matrices in order
// (log-semiring), sums gold partials, emits [gold, logZ].
// ---------------------------------------------------------------------------
__global__ __launch_bounds__(256) void crf_reduce_kernel(
    const float* __restrict__ ws,     // [NCHUNK,5,B]
    float*       __restrict__ out)    // [2,B]
{
    const int b = blockIdx.x * blockDim.x + threadIdx.x;
    float c00 = 0.0f, c01 = NEGI, c10 = NEGI, c11 = 0.0f;   // identity
    float gold = 0.0f;
    #pragma unroll 4
    for (int c = 0; c < NCHUNK; ++c) {
        const float* p = ws + (size_t)c * 5 * BB;
        float l00 = p[0 * BB + b];
        float l01 = p[1 * BB + b];
        float l10 = p[2 * BB + b];
        float l11 = p[3 * BB + b];
        gold     += p[4 * BB + b];
        float n00 = lse2(c00 + l00, c01 + l10);
        float n01 = lse2(c00 + l01, c01 + l11);
        float n10 = lse2(c10 + l00, c11 + l10);
        float n11 = lse2(c10 + l01, c11 + l11);
        c00 = n00; c01 = n01; c10 = n10; c11 = n11;
    }
    // alpha0 = (0,0): alpha_S[j] = lse_i(C[i][j]); logZ = lse_j alpha_S[j]
    float s0 = lse2(c00, c10);
    float s1 = lse2(c01, c11);
    out[b]      = gold;              // row 0: gold scores
    out[BB + b] = lse2(s0, s1);      // row 1: log partition
}

extern "C" void kernel_launch(void* const* d_in, const int* in_sizes, int n_in,
                              void* d_out, int out_size, void* d_ws, size_t ws_size,
                              hipStream_t stream) {
    const float* em   = (const float*)d_in[0];
    const float* tp   = (const float*)d_in[1];
    const int*   tags = (const int*)d_in[2];
    const int*   w2w  = (const int*)d_in[3];
    const int*   itm  = (const int*)d_in[4];
    const int*   dst  = (const int*)d_in[5];
    float* out = (float*)d_out;
    float* ws  = (float*)d_ws;

    dim3 grid1(BB / 256, NCHUNK);
    crf_chunk_kernel<<<grid1, 256, 0, stream>>>(em, tp, tags, w2w, itm, dst, ws);
    crf_reduce_kernel<<<BB / 256, 256, 0, stream>>>(ws, out);
}